// GATLayer_76931454206117
// MI455X (gfx1250) — compile-verified
//
#include <hip/hip_runtime.h>
#include <hip/hip_bf16.h>

// Problem constants (from reference setup_inputs)
#define B   128
#define N   64
#define P   4032
#define FIN 128
#define FOUT 128
#define ROWS (B * N)          // 8192 rows of h
#define AROW 258              // a_input row length: 2 + 128 + 128
#define ALPHA 0.2f

typedef float v2f __attribute__((ext_vector_type(2)));
typedef float v8f __attribute__((ext_vector_type(8)));

// ---------------------------------------------------------------------------
// Kernel 1: h = input @ W  via V_WMMA_F32_16X16X4_F32
// input: (ROWS, FIN) row-major, W: (FIN, FOUT) row-major, h: (ROWS, FOUT)
// One wave computes one 16x16 tile of h; 8 waves/block cover all 128 cols.
// A frag (16x4 f32): lanes 0-15 hold K=0,1 ; lanes 16-31 hold K=2,3 (per lane row M)
// B frag (4x16 f32): mirrored — lanes hold column N, VGPRs hold K pairs
// C/D (16x16 f32, 8 VGPRs): VGPR r -> M = r + 8*(lane>=16), N = lane&15
// ---------------------------------------------------------------------------
__global__ void __launch_bounds__(256) gat_wmma_h_kernel(
    const float* __restrict__ input, const float* __restrict__ W,
    float* __restrict__ h) {
  const int wave = threadIdx.x >> 5;        // 0..7 -> N tile
  const int lane = threadIdx.x & 31;
  const int m0   = blockIdx.x * 16;         // 512 blocks cover 8192 rows
  const int n0   = wave * 16;
  const int mc   = lane & 15;               // row for A, col for B/C
  const int half = lane >> 4;               // 0 or 1
  const int kb   = half * 2;                // K offset within 4-wide slab

  v8f c = {};
  const float* Arow = input + (size_t)(m0 + mc) * FIN;
  for (int k0 = 0; k0 < FIN; k0 += 4) {
    v2f a, b;
    a.x = Arow[k0 + kb];
    a.y = Arow[k0 + kb + 1];
    b.x = W[(size_t)(k0 + kb)     * FOUT + n0 + mc];
    b.y = W[(size_t)(k0 + kb + 1) * FOUT + n0 + mc];
    // (neg_a, A, neg_b, B, c_mod, C, reuse_a, reuse_b)
    c = __builtin_amdgcn_wmma_f32_16x16x4_f32(false, a, false, b,
                                              (short)0, c, false, false);
  }
#pragma unroll
  for (int r = 0; r < 8; ++r) {
    const int m = m0 + r + half * 8;
    h[(size_t)m * FOUT + n0 + mc] = c[r];
  }
}

// ---------------------------------------------------------------------------
// Kernel 2: fold attention vector into per-(b,n) scalars:
//   s1[i] = h[i,:] . a[2:130],  s2[i] = h[i,:] . a[130:258]
// ---------------------------------------------------------------------------
__global__ void __launch_bounds__(256) gat_s12_kernel(
    const float* __restrict__ h, const float* __restrict__ a,
    float* __restrict__ s1, float* __restrict__ s2) {
  const int i = blockIdx.x * blockDim.x + threadIdx.x;
  if (i >= ROWS) return;
  const float* hr = h + (size_t)i * FOUT;
  float acc1 = 0.f, acc2 = 0.f;
#pragma unroll 4
  for (int f = 0; f < FOUT; ++f) {
    const float v = hr[f];
    acc1 += v * a[2 + f];
    acc2 += v * a[130 + f];
  }
  s1[i] = acc1;
  s2[i] = acc2;
}

// ---------------------------------------------------------------------------
// Kernel 3: the bandwidth kernel. For each row (b,p):
//   a_input[b,p,:] = [ g[b,pid[p],0:2] , h[b,e0,:] , h[b,e1,:] ]   (258 floats)
//   e[b,p] = leaky_relu(a0*g0 + a1*g1 + s1[b,e0] + s2[b,e1])
// One 256-thread block per row iteration; writes are fully coalesced,
// h/g reads hit L2 (h is 4MB, g is 4MB; global L2 = 192MB).
// ---------------------------------------------------------------------------
__global__ void __launch_bounds__(256) gat_gather_kernel(
    const float* __restrict__ g, const int* __restrict__ pid,
    const int* __restrict__ edges, const float* __restrict__ h,
    const float* __restrict__ s1, const float* __restrict__ s2,
    const float* __restrict__ avec,
    float* __restrict__ a_input, float* __restrict__ e_out) {
  const float a0 = avec[0];
  const float a1 = avec[1];
  const int t = threadIdx.x;

  for (int row = blockIdx.x; row < B * P; row += gridDim.x) {
    const int b = row / P;
    const int p = row - b * P;
    const int gp = pid[p];                  // wave-uniform -> scalar loads
    const int e0 = edges[p * 2 + 0];
    const int e1 = edges[p * 2 + 1];

    const float* gsrc = g + ((size_t)b * P + gp) * 2;
    const float* hs   = h + ((size_t)b * N + e0) * FOUT;
    const float* hd   = h + ((size_t)b * N + e1) * FOUT;
    float* dst = a_input + (size_t)row * AROW;

    // elements 0..255 by thread id, 256..257 by threads 0,1 (second pass)
    float v;
    if (t < 2)        v = gsrc[t];
    else if (t < 130) v = hs[t - 2];
    else              v = hd[t - 130];
    dst[t] = v;
    if (t < 2) dst[256 + t] = hd[126 + t];

    if (t == 0) {
      float ev = a0 * gsrc[0] + a1 * gsrc[1]
               + s1[b * N + e0] + s2[b * N + e1];
      e_out[row] = (ev > 0.f) ? ev : ALPHA * ev;
    }
  }
}

// ---------------------------------------------------------------------------
// Kernel 4: softmax over P per batch row. One block per b.
// ---------------------------------------------------------------------------
__global__ void __launch_bounds__(256) gat_softmax_kernel(
    const float* __restrict__ e, float* __restrict__ att) {
  __shared__ float red[256];
  const int b = blockIdx.x;
  const int t = threadIdx.x;
  const float* er = e + (size_t)b * P;

  float m = -3.402823466e38f;
  for (int p = t; p < P; p += 256) m = fmaxf(m, er[p]);
  red[t] = m;
  __syncthreads();
  for (int s = 128; s > 0; s >>= 1) {
    if (t < s) red[t] = fmaxf(red[t], red[t + s]);
    __syncthreads();
  }
  m = red[0];
  __syncthreads();

  float sum = 0.f;
  for (int p = t; p < P; p += 256) sum += __expf(er[p] - m);
  red[t] = sum;
  __syncthreads();
  for (int s = 128; s > 0; s >>= 1) {
    if (t < s) red[t] += red[t + s];
    __syncthreads();
  }
  const float inv = 1.0f / red[0];

  for (int p = t; p < P; p += 256)
    att[(size_t)b * P + p] = __expf(er[p] - m) * inv;
}

// ---------------------------------------------------------------------------
extern "C" void kernel_launch(void* const* d_in, const int* in_sizes, int n_in,
                              void* d_out, int out_size, void* d_ws, size_t ws_size,
                              hipStream_t stream) {
  const float* input = (const float*)d_in[0];
  // d_in[1] = ag (unused by reference)
  const float* g     = (const float*)d_in[2];
  const int*   pid   = (const int*)d_in[3];
  const int*   edges = (const int*)d_in[4];
  // d_in[5] = n_permutations (compile-time constant P)
  const float* W     = (const float*)d_in[6];
  const float* avec  = (const float*)d_in[7];

  float* out     = (float*)d_out;
  float* att     = out;                 // (B, P)
  float* a_input = out + (size_t)B * P; // (B, P, 258)

  char* ws = (char*)d_ws;
  float* h  = (float*)(ws);                                   // ROWS*FOUT*4 = 4 MB
  float* s1 = (float*)(ws + (size_t)ROWS * FOUT * 4);         // 32 KB
  float* s2 = (float*)(ws + (size_t)ROWS * FOUT * 4 + 32768); // 32 KB
  float* e  = (float*)(ws + (size_t)ROWS * FOUT * 4 + 65536); // B*P*4 = 2 MB

  gat_wmma_h_kernel<<<dim3(ROWS / 16), dim3(256), 0, stream>>>(input, W, h);
  gat_s12_kernel<<<dim3(ROWS / 256), dim3(256), 0, stream>>>(h, avec, s1, s2);
  gat_gather_kernel<<<dim3(8192), dim3(256), 0, stream>>>(g, pid, edges, h, s1, s2,
                                                          avec, a_input, e);
  gat_softmax_kernel<<<dim3(B), dim3(256), 0, stream>>>(e, att);
}